// RSNN_10995116278570
// MI455X (gfx1250) — compile-verified
//
#include <hip/hip_runtime.h>
#include <math.h>

// ---------------------------------------------------------------------------
// RSNN step for MI455X (gfx1250, wave32).
//
//   out = [ new_z (16x2048) | new_v (16x2048) | new_z2 (16x2048) ]
//
// GEMM1: inputs(16x512) @ input_weights(512x2048)   -> fused LIF epilogue
// GEMM2: new_z(16x2048) @ gathered bernoulli columns (2048 x 2048-of-28672)
//        (never materializes the 16x28672 product; 14x less compute,
//         ~16.8 MB of useful weight data instead of 235 MB)
//
// Both GEMMs use V_WMMA_F32_16X16X4_F32 (full fp32, exact vs reference up to
// summation order). ISA lane layout (CDNA5 ISA §7.12.2):
//   A (16x4):  lanes 0-15 : VGPR0=K0, VGPR1=K1 ; lanes 16-31 : K2, K3 (M=lane&15)
//   B (4x16):  lanes 0-15 : VGPR0=K0 row, VGPR1=K1 row ; lanes 16-31: K2,K3 (N=lane&15)
//   C/D:       acc[i] = D[m][n], m = i + 8*(lane>>4), n = lane&15
//
// Block = 256 threads = 8 waves = 2 N-tiles x 4 K-splits, ordered LDS reduce
// (deterministic summation order; 512 waves total for memory-level
//  parallelism on the scattered bernoulli column loads).
// ---------------------------------------------------------------------------

typedef __attribute__((ext_vector_type(2))) float v2f;
typedef __attribute__((ext_vector_type(8))) float v8f;

#define THR_F 0.03f
#define DECAY 0.9512294245007140f          // exp(-1/20)
#define ONE_MINUS_DECAY 0.0487705754992860f
#define TWO_PI 6.2831853071795864769f

// ---------------------------------------------------------------------------
// Kernel 1: i_in GEMM + LIF update + spike.
//   grid = 64 blocks, block = 256.  Each block: 2 N-tiles of 16, K=512 split 4x128.
// ---------------------------------------------------------------------------
__global__ __launch_bounds__(256) void rsnn_step1(
    const float* __restrict__ inputs,        // 16 x 512
    const float* __restrict__ old_v,         // 16 x 2048
    const float* __restrict__ old_spike,     // 16 x 2048
    const float* __restrict__ input_weights, // 512 x 2048
    const int*   __restrict__ tt_p,
    float* __restrict__ out_z,               // 16 x 2048
    float* __restrict__ out_v)               // 16 x 2048
{
    __shared__ float red[8 * 256];           // per-wave v8f partials

    const int lane = threadIdx.x & 31;
    const int wave = threadIdx.x >> 5;
    const int tile = wave >> 2;              // 0..1   (N tile within block)
    const int ksp  = wave & 3;               // 0..3   (K split)
    const int hsel = lane >> 4;              // 0/1 -> K pair base 0 / 2
    const int ml   = lane & 15;              // A row / B,D column in tile
    const int n    = (blockIdx.x * 2 + tile) * 16 + ml;

    const float* arow = inputs + ml * 512 + ksp * 128 + 2 * hsel;
    const float* wcol = input_weights + n + (size_t)(ksp * 128 + 2 * hsel) * 2048;

    v8f acc = {};
    #pragma unroll 8
    for (int k0 = 0; k0 < 128; k0 += 4) {
        v2f a = *(const v2f*)(arow + k0);                 // A[ml][k], A[ml][k+1]
        v2f b;
        b.x = wcol[(size_t)k0 * 2048];                    // B[k  ][n]
        b.y = wcol[(size_t)k0 * 2048 + 2048];             // B[k+1][n]
        acc = __builtin_amdgcn_wmma_f32_16x16x4_f32(
            false, a, false, b, (short)0, acc, false, false);
    }

    #pragma unroll
    for (int i = 0; i < 8; ++i)
        red[wave * 256 + i * 32 + lane] = acc[i];
    __syncthreads();

    if (ksp == 0) {                                       // waves 0 and 4 reduce
        // Fast hardware sin (v_sin_f32): argument is tiny, bias needs no
        // Payne-Hanek slow path.
        const float bias =
            0.5f * __sinf(TWO_PI * 4.0f * (0.001f * (float)(*tt_p)));
        const int base = tile * 4 * 256;
        #pragma unroll
        for (int i = 0; i < 8; ++i) {
            const int off = i * 32 + lane;
            float s = red[base + off];                    // ordered, deterministic
            s += red[base + 256 + off];
            s += red[base + 512 + off];
            s += red[base + 768 + off];

            const int m   = i + 8 * hsel;
            const int idx = m * 2048 + n;
            const float i_in = s + bias;
            const float nv = DECAY * old_v[idx] + ONE_MINUS_DECAY * i_in
                             - THR_F * old_spike[idx];
            const float vs = (nv - THR_F) * (1.0f / THR_F);
            out_z[idx] = (vs > 0.0f) ? 1.0f : 0.0f;
            out_v[idx] = nv;
        }
    }
}

// ---------------------------------------------------------------------------
// Kernel 2: gathered GEMM  z(16x2048) @ W[:, random_index]  -> new_z2.
//   grid = 64 blocks, block = 256.  K=2048 split 4x512 per block.
// ---------------------------------------------------------------------------
__global__ __launch_bounds__(256) void rsnn_step2(
    const float* __restrict__ z,       // 16 x 2048 (kernel 1 output)
    const float* __restrict__ bern,    // 2048 x 28672 (row-major)
    const int*   __restrict__ ridx,    // 2048
    float* __restrict__ out_z2)        // 16 x 2048
{
    __shared__ float red[8 * 256];

    const int lane = threadIdx.x & 31;
    const int wave = threadIdx.x >> 5;
    const int tile = wave >> 2;
    const int ksp  = wave & 3;
    const int hsel = lane >> 4;
    const int ml   = lane & 15;
    const int n    = (blockIdx.x * 2 + tile) * 16 + ml;
    const int col  = ridx[n];                              // gathered column

    const float* zrow = z + ml * 2048 + ksp * 512 + 2 * hsel;
    const float* wcol = bern + (size_t)(ksp * 512 + 2 * hsel) * 28672 + col;

    v8f acc = {};
    #pragma unroll 8
    for (int k0 = 0; k0 < 512; k0 += 4) {
        v2f a = *(const v2f*)(zrow + k0);                  // Z[ml][k], Z[ml][k+1]
        v2f b;
        b.x = wcol[(size_t)k0 * 28672];                    // W[k  ][col]
        b.y = wcol[(size_t)k0 * 28672 + 28672];            // W[k+1][col]
        acc = __builtin_amdgcn_wmma_f32_16x16x4_f32(
            false, a, false, b, (short)0, acc, false, false);
    }

    #pragma unroll
    for (int i = 0; i < 8; ++i)
        red[wave * 256 + i * 32 + lane] = acc[i];
    __syncthreads();

    if (ksp == 0) {
        const int base = tile * 4 * 256;
        #pragma unroll
        for (int i = 0; i < 8; ++i) {
            const int off = i * 32 + lane;
            float s = red[base + off];
            s += red[base + 256 + off];
            s += red[base + 512 + off];
            s += red[base + 768 + off];
            const int m = i + 8 * hsel;
            out_z2[m * 2048 + n] = s;
        }
    }
}

// ---------------------------------------------------------------------------
extern "C" void kernel_launch(void* const* d_in, const int* in_sizes, int n_in,
                              void* d_out, int out_size, void* d_ws, size_t ws_size,
                              hipStream_t stream) {
    const float* inputs        = (const float*)d_in[0];   // 16*512
    const float* old_v         = (const float*)d_in[1];   // 16*2048
    const float* old_spike     = (const float*)d_in[2];   // 16*2048
    const float* input_weights = (const float*)d_in[3];   // 512*2048
    const float* bernoulli     = (const float*)d_in[4];   // 2048*28672
    const int*   random_index  = (const int*)d_in[5];     // 2048
    const int*   tt            = (const int*)d_in[6];     // 1

    float* out    = (float*)d_out;
    float* out_z  = out;                  // 16*2048
    float* out_v  = out + 16 * 2048;      // 16*2048
    float* out_z2 = out + 2 * 16 * 2048;  // 16*2048

    (void)in_sizes; (void)n_in; (void)out_size; (void)d_ws; (void)ws_size;

    dim3 blk(256);
    // 128 N-tiles / 2 tiles per block = 64 blocks, 8 waves each = 512 waves.
    rsnn_step1<<<dim3(64), blk, 0, stream>>>(inputs, old_v, old_spike,
                                             input_weights, tt, out_z, out_v);
    rsnn_step2<<<dim3(64), blk, 0, stream>>>(out_z, bernoulli, random_index,
                                             out_z2);
}